// GNN_16724602650757
// MI455X (gfx1250) — compile-verified
//
#include <hip/hip_runtime.h>

typedef __attribute__((ext_vector_type(2))) float v2f;
typedef __attribute__((ext_vector_type(8))) float v8f;

#define N_NODES 100000
#define N_EDGES 3200000

// ---------------- degree / normalization precompute ----------------

__global__ void k_init_deg(float* __restrict__ deg) {
    int i = blockIdx.x * blockDim.x + threadIdx.x;
    if (i < N_NODES) deg[i] = 1.0f;               // self-loop counts once
}

__global__ void k_count_deg(const int* __restrict__ dst, float* __restrict__ deg) {
    int e = blockIdx.x * blockDim.x + threadIdx.x;
    if (e < N_EDGES) atomicAdd(&deg[dst[e]], 1.0f);
}

__global__ void k_to_dis(float* __restrict__ deg) {
    int i = blockIdx.x * blockDim.x + threadIdx.x;
    if (i < N_NODES) deg[i] = rsqrtf(deg[i]);     // deg >= 1 always
}

// ---------------- f32 WMMA GEMM: Y[nrows x OUT] = X[nrows x IN] @ W[IN x OUT] ----------------
// One wave32 per 16-row tile; V_WMMA_F32_16X16X4_F32, K stepped by 4.

template <int IN, int OUT>
__global__ __launch_bounds__(256) void k_gemm_wmma(const float* __restrict__ X,
                                                   const float* __restrict__ W,
                                                   float* __restrict__ Y,
                                                   int nrows) {
    const int lane = threadIdx.x & 31;
    const int wave = blockIdx.x * (blockDim.x >> 5) + (threadIdx.x >> 5);
    const int row0 = wave * 16;
    if (row0 >= nrows) return;                    // wave-uniform: EXEC stays all-1s

    const int half = lane >> 4;                   // 0: lanes 0-15, 1: lanes 16-31
    const int l15  = lane & 15;
    const int arow = row0 + l15;

    constexpr int NT = OUT / 16;                  // number of 16-wide N tiles
    v8f acc[NT];
    #pragma unroll
    for (int t = 0; t < NT; ++t)
        #pragma unroll
        for (int v = 0; v < 8; ++v) acc[t][v] = 0.0f;

    for (int k = 0; k < IN; k += 4) {
        // A fragment (16x4): lane half selects K pair {k,k+1} or {k+2,k+3}
        const float* ap = X + (size_t)arow * IN + (k + 2 * half);
        v2f a;
        a.x = ap[0];
        a.y = ap[1];

        #pragma unroll
        for (int t = 0; t < NT; ++t) {
            // B fragment (4x16): VGPR0 = rows {k, k+2}, VGPR1 = rows {k+1, k+3}
            const int col = t * 16 + l15;
            v2f b;
            b.x = W[(k + 2 * half)     * OUT + col];
            b.y = W[(k + 2 * half + 1) * OUT + col];
            acc[t] = __builtin_amdgcn_wmma_f32_16x16x4_f32(
                /*neg_a=*/false, a, /*neg_b=*/false, b,
                /*c_mod=*/(short)0, acc[t], /*reuse_a=*/false, /*reuse_b=*/false);
        }
    }

    // C/D layout: VGPR v -> row = v + 8*half, col = lane%16 (+16*t)
    #pragma unroll
    for (int t = 0; t < NT; ++t)
        #pragma unroll
        for (int v = 0; v < 8; ++v) {
            int r = row0 + v + 8 * half;
            int c = t * 16 + l15;
            Y[(size_t)r * OUT + c] = acc[t][v];
        }
}

// ---------------- aggregation: out[dst] += dis[src]*dis[dst]*hw[src], self-loop, bias/relu ----

template <int OUT>
__global__ void k_self_init(const float* __restrict__ hw, const float* __restrict__ dis,
                            float* __restrict__ agg) {
    long long t = (long long)blockIdx.x * blockDim.x + threadIdx.x;
    if (t >= (long long)N_NODES * OUT) return;
    int i = (int)(t / OUT);
    float d = dis[i];
    agg[t] = d * d * hw[t];                       // initializes buffer (no memset needed)
}

template <int OUT>
__global__ void k_edge_agg(const int* __restrict__ src, const int* __restrict__ dst,
                           const float* __restrict__ dis, const float* __restrict__ hw,
                           float* __restrict__ agg) {
    long long t = (long long)blockIdx.x * blockDim.x + threadIdx.x;
    if (t >= (long long)N_EDGES * OUT) return;
    int e = (int)(t / OUT);
    int f = (int)(t % OUT);                       // wave covers contiguous features of 32/OUT edges
    int s = src[e];
    int d = dst[e];
    float nrm = dis[s] * dis[d];
    atomicAdd(&agg[(long long)d * OUT + f], nrm * hw[(long long)s * OUT + f]);
}

template <int OUT>
__global__ void k_bias_relu(float* __restrict__ h, const float* __restrict__ b) {
    long long t = (long long)blockIdx.x * blockDim.x + threadIdx.x;
    if (t >= (long long)N_NODES * OUT) return;
    float v = h[t] + b[t % OUT];
    h[t] = v > 0.0f ? v : 0.0f;
}

// ---------------- layer 3 (out = 1): VALU dot product + scalar scatter ----------------

__global__ void k_gemm_out1(const float* __restrict__ h2, const float* __restrict__ W3,
                            float* __restrict__ hw3) {
    int i = blockIdx.x * blockDim.x + threadIdx.x;
    if (i >= N_NODES) return;
    float s = 0.0f;
    #pragma unroll
    for (int k = 0; k < 16; ++k) s += h2[(size_t)i * 16 + k] * W3[k];
    hw3[i] = s;
}

__global__ void k_self_init_out1(const float* __restrict__ hw3, const float* __restrict__ dis,
                                 const float* __restrict__ b3, float* __restrict__ out) {
    int i = blockIdx.x * blockDim.x + threadIdx.x;
    if (i >= N_NODES) return;
    float d = dis[i];
    out[i] = d * d * hw3[i] + b3[0];              // bias folded into init (commutes with adds)
}

__global__ void k_edge_agg_out1(const int* __restrict__ src, const int* __restrict__ dst,
                                const float* __restrict__ dis, const float* __restrict__ hw3,
                                float* __restrict__ out) {
    int e = blockIdx.x * blockDim.x + threadIdx.x;
    if (e >= N_EDGES) return;
    int s = src[e];
    int d = dst[e];
    atomicAdd(&out[d], dis[s] * dis[d] * hw3[s]);
}

// ---------------- launcher ----------------

static inline int blks(long long n, int bs) { return (int)((n + bs - 1) / bs); }

extern "C" void kernel_launch(void* const* d_in, const int* in_sizes, int n_in,
                              void* d_out, int out_size, void* d_ws, size_t ws_size,
                              hipStream_t stream) {
    (void)in_sizes; (void)n_in; (void)out_size; (void)ws_size;

    const float* x   = (const float*)d_in[0];     // [N,128]
    const int*   ei  = (const int*)  d_in[1];     // [2,E]
    const float* W1  = (const float*)d_in[2];     // [128,32]
    const float* b1  = (const float*)d_in[3];
    const float* W2  = (const float*)d_in[4];     // [32,16]
    const float* b2  = (const float*)d_in[5];
    const float* W3  = (const float*)d_in[6];     // [16,1]
    const float* b3  = (const float*)d_in[7];
    float* out = (float*)d_out;                   // [N,1]

    const int* src = ei;
    const int* dst = ei + N_EDGES;

    // workspace layout (floats): all buffers L2-resident (~39 MB total)
    float* ws   = (float*)d_ws;
    float* dis  = ws;                      // N
    float* hw1  = dis  + N_NODES;          // 32N
    float* h1   = hw1  + 32 * N_NODES;     // 32N (agg1, then relu'd in place)
    float* hw2  = h1   + 32 * N_NODES;     // 16N
    float* h2   = hw2  + 16 * N_NODES;     // 16N
    float* hw3  = h2   + 16 * N_NODES;     // N

    const int BS = 256;

    // deg -> dis = rsqrt(deg)
    k_init_deg <<<blks(N_NODES, BS), BS, 0, stream>>>(dis);
    k_count_deg<<<blks(N_EDGES, BS), BS, 0, stream>>>(dst, dis);
    k_to_dis   <<<blks(N_NODES, BS), BS, 0, stream>>>(dis);

    const int gemm_grid = blks(N_NODES / 16, BS / 32); // waves of 16 rows, 8 waves/block

    // layer 1: 128 -> 32
    k_gemm_wmma<128, 32><<<gemm_grid, BS, 0, stream>>>(x, W1, hw1, N_NODES);
    k_self_init<32><<<blks((long long)N_NODES * 32, BS), BS, 0, stream>>>(hw1, dis, h1);
    k_edge_agg <32><<<blks((long long)N_EDGES * 32, BS), BS, 0, stream>>>(src, dst, dis, hw1, h1);
    k_bias_relu<32><<<blks((long long)N_NODES * 32, BS), BS, 0, stream>>>(h1, b1);

    // layer 2: 32 -> 16
    k_gemm_wmma<32, 16><<<gemm_grid, BS, 0, stream>>>(h1, W2, hw2, N_NODES);
    k_self_init<16><<<blks((long long)N_NODES * 16, BS), BS, 0, stream>>>(hw2, dis, h2);
    k_edge_agg <16><<<blks((long long)N_EDGES * 16, BS), BS, 0, stream>>>(src, dst, dis, hw2, h2);
    k_bias_relu<16><<<blks((long long)N_NODES * 16, BS), BS, 0, stream>>>(h2, b2);

    // layer 3: 16 -> 1 (no relu)
    k_gemm_out1     <<<blks(N_NODES, BS), BS, 0, stream>>>(h2, W3, hw3);
    k_self_init_out1<<<blks(N_NODES, BS), BS, 0, stream>>>(hw3, dis, b3, out);
    k_edge_agg_out1 <<<blks(N_EDGES, BS), BS, 0, stream>>>(src, dst, dis, hw3, out);
}